// GTGModule_59682865545589
// MI455X (gfx1250) — compile-verified
//
#include <hip/hip_runtime.h>
#include <hip/hip_bf16.h>
#include <math.h>

typedef float v2f __attribute__((ext_vector_type(2)));
typedef float v8f __attribute__((ext_vector_type(8)));

#define BSZ   4096
#define DIM   1024
#define NCLS  100
#define CPAD  128
#define NLAB  2048
#define ITERS 30

// workspace layout (float offsets)
#define NE_OFF  0u                       // 4096*1024
#define A_OFF   (4096u*1024u)            // 4096*4096
#define X_OFF   (A_OFF + 4096u*4096u)    // 4096*128
#define AX_OFF  (X_OFF + 4096u*128u)     // 4096*128
#define SUM_OFF (AX_OFF + 4096u*128u)    // 1

// ---------------------------------------------------------------- init X0 + zero mean-accumulator
__global__ void __launch_bounds__(256) init_x(const int* __restrict__ lab,
                                              float* __restrict__ X,
                                              float* __restrict__ sum) {
  int i = blockIdx.x * 256 + threadIdx.x;
  if (i == 0) *sum = 0.0f;
  if (i < BSZ * CPAD) {
    int b = i >> 7, c = i & 127;
    float v;
    if (c >= NCLS)      v = 0.0f;
    else if (b >= NLAB) v = 1.0f / NCLS;
    else                v = (lab[b] == c) ? 1.0f : 0.0f;
    X[i] = v;
  }
}

// ---------------------------------------------------------------- row L2-normalize
__global__ void __launch_bounds__(256) normalize_rows(const float* __restrict__ emb,
                                                      float* __restrict__ ne) {
  int row = blockIdx.x;
  const float* e = emb + (size_t)row * DIM;
  float ss = 0.0f;
  for (int i = threadIdx.x; i < DIM; i += 256) { float v = e[i]; ss += v * v; }
  __shared__ float red[256];
  red[threadIdx.x] = ss;
  __syncthreads();
  for (int s = 128; s > 0; s >>= 1) {
    if (threadIdx.x < s) red[threadIdx.x] += red[threadIdx.x + s];
    __syncthreads();
  }
  float inv = 1.0f / (sqrtf(red[0]) + 1e-12f);
  float* o = ne + (size_t)row * DIM;
  for (int i = threadIdx.x; i < DIM; i += 256) o[i] = e[i] * inv;
}

// ---------------------------------------------------------------- A = ne * ne^T (fused diag-zero, clamp, sum)
// block: 256 thr (8 waves). tile: 64 rows x 128 cols.
// wave w: row-group wr = w>>1 (16 rows), col-group wc = w&1 (64 cols = 4 tiles).
__global__ void __launch_bounds__(256) gemm_syrk(const float* __restrict__ ne,
                                                 float* __restrict__ A,
                                                 float* __restrict__ sum) {
  __shared__ float rowT[64 * 32];
  __shared__ float colT[128 * 32];
  __shared__ float red[256];
  const int r0 = blockIdx.y * 64;
  const int c0 = blockIdx.x * 128;
  const int t = threadIdx.x, w = t >> 5, l = t & 31;
  const int wr = w >> 1, wc = w & 1;
  const int mrow = l & 15;
  const int khi  = (l >> 4) * 2;             // K sub-pair select per ISA 16x16x4 layout

  v8f acc0 = {0.f,0.f,0.f,0.f,0.f,0.f,0.f,0.f};
  v8f acc1 = acc0, acc2 = acc0, acc3 = acc0;

  for (int k0 = 0; k0 < DIM; k0 += 32) {
    #pragma unroll
    for (int j = 0; j < 4; ++j) { // 64x32 row tile (2048 floats, 8/thread)
      int i = (t + 256 * j) * 2, r = i >> 5, k = i & 31;
      *(v2f*)(rowT + i) = *(const v2f*)(ne + (size_t)(r0 + r) * DIM + k0 + k);
    }
    #pragma unroll
    for (int j = 0; j < 8; ++j) { // 128x32 col tile (4096 floats, 16/thread)
      int i = (t + 256 * j) * 2, r = i >> 5, k = i & 31;
      *(v2f*)(colT + i) = *(const v2f*)(ne + (size_t)(c0 + r) * DIM + k0 + k);
    }
    __syncthreads();
    const float* ra = rowT + (wr * 16 + mrow) * 32 + khi;
    const float* rb = colT + (wc * 64 + mrow) * 32 + khi;
    #pragma unroll
    for (int kk = 0; kk < 32; kk += 4) {
      v2f a  = *(const v2f*)(ra + kk);
      v2f b0 = *(const v2f*)(rb + kk);
      v2f b1 = *(const v2f*)(rb + 16 * 32 + kk);
      v2f b2 = *(const v2f*)(rb + 32 * 32 + kk);
      v2f b3 = *(const v2f*)(rb + 48 * 32 + kk);
      acc0 = __builtin_amdgcn_wmma_f32_16x16x4_f32(false, a, false, b0, (short)0, acc0, false, false);
      acc1 = __builtin_amdgcn_wmma_f32_16x16x4_f32(false, a, false, b1, (short)0, acc1, false, false);
      acc2 = __builtin_amdgcn_wmma_f32_16x16x4_f32(false, a, false, b2, (short)0, acc2, false, false);
      acc3 = __builtin_amdgcn_wmma_f32_16x16x4_f32(false, a, false, b3, (short)0, acc3, false, false);
    }
    __syncthreads();
  }

  // epilogue: D VGPR v -> row +v + 8*(l>>4); col tile j at wc*64 + 16j
  float part = 0.0f;
  const int rbase = r0 + wr * 16 + ((l >> 4) ? 8 : 0);
  v8f accs[4] = {acc0, acc1, acc2, acc3};
  #pragma unroll
  for (int j = 0; j < 4; ++j) {
    const int gc = c0 + wc * 64 + j * 16 + (l & 15);
    #pragma unroll
    for (int v = 0; v < 8; ++v) {
      int gr = rbase + v;
      float val = accs[j][v];
      if (gr == gc) val = 0.0f;
      val = fminf(fmaxf(val, 0.0f), 1.0f);
      A[(size_t)gr * BSZ + gc] = val;
      part += val;
    }
  }
  red[t] = part;
  __syncthreads();
  for (int s = 128; s > 0; s >>= 1) {
    if (t < s) red[t] += red[t + s];
    __syncthreads();
  }
  if (t == 0) atomicAdd(sum, red[0]);
}

// ---------------------------------------------------------------- A = (A < mean) ? 0 : A
__global__ void __launch_bounds__(256) threshold_mean(float* __restrict__ A,
                                                      const float* __restrict__ sum) {
  float mean = *sum * (1.0f / ((float)BSZ * (float)BSZ));
  size_t i = ((size_t)blockIdx.x * 256 + threadIdx.x) * 4;
  #pragma unroll
  for (int j = 0; j < 4; ++j) {
    float v = A[i + j];
    A[i + j] = (v < mean) ? 0.0f : v;
  }
}

// ---------------------------------------------------------------- AX = A @ X  (M=4096,K=4096,N=128pad)
// block tile: 32 rows x 128 cols. wave w: wr = w>>2 (16 rows), wc = w&3 (32 cols = 2 tiles).
__global__ void __launch_bounds__(256) gemm_ax(const float* __restrict__ A,
                                               const float* __restrict__ X,
                                               float* __restrict__ AX) {
  __shared__ float At[32 * 32];
  __shared__ float Xt[32 * 128];
  const int r0 = blockIdx.x * 32;
  const int t = threadIdx.x, w = t >> 5, l = t & 31;
  const int wr = w >> 2, wc = w & 3;
  const int mrow = l & 15;
  const int khi  = (l >> 4) * 2;

  v8f acc0 = {0.f,0.f,0.f,0.f,0.f,0.f,0.f,0.f};
  v8f acc1 = acc0;

  for (int k0 = 0; k0 < BSZ; k0 += 32) {
    #pragma unroll
    for (int j = 0; j < 2; ++j) { // 32x32 A tile (1024 floats, 4/thread)
      int i = (t + 256 * j) * 2, r = i >> 5, k = i & 31;
      *(v2f*)(At + i) = *(const v2f*)(A + (size_t)(r0 + r) * BSZ + k0 + k);
    }
    #pragma unroll
    for (int j = 0; j < 8; ++j) { // 32x128 X tile: rows contiguous in X
      int i = (t + 256 * j) * 2;
      *(v2f*)(Xt + i) = *(const v2f*)(X + (size_t)k0 * CPAD + i);
    }
    __syncthreads();
    const float* ra = At + (wr * 16 + mrow) * 32 + khi;
    const int    cb = wc * 32 + mrow;
    #pragma unroll
    for (int kk = 0; kk < 32; kk += 4) {
      v2f a = *(const v2f*)(ra + kk);
      v2f b0, b1;
      b0.x = Xt[(kk + khi)     * CPAD + cb];
      b0.y = Xt[(kk + khi + 1) * CPAD + cb];
      b1.x = Xt[(kk + khi)     * CPAD + cb + 16];
      b1.y = Xt[(kk + khi + 1) * CPAD + cb + 16];
      acc0 = __builtin_amdgcn_wmma_f32_16x16x4_f32(false, a, false, b0, (short)0, acc0, false, false);
      acc1 = __builtin_amdgcn_wmma_f32_16x16x4_f32(false, a, false, b1, (short)0, acc1, false, false);
    }
    __syncthreads();
  }
  const int rbase = r0 + wr * 16 + ((l >> 4) ? 8 : 0);
  const int gc0   = wc * 32 + (l & 15);
  #pragma unroll
  for (int v = 0; v < 8; ++v) {
    AX[(size_t)(rbase + v) * CPAD + gc0]      = acc0[v];
    AX[(size_t)(rbase + v) * CPAD + gc0 + 16] = acc1[v];
  }
}

// ---------------------------------------------------------------- replicator normalize + entropy + X update
// one wave per row; lane l owns cols 4l..4l+3
__global__ void __launch_bounds__(256) replicator_step(const float* __restrict__ AX,
                                                       float* __restrict__ X,
                                                       float* __restrict__ outX,
                                                       float* __restrict__ outEnt,
                                                       int it) {
  const int t = threadIdx.x, w = t >> 5, l = t & 31;
  const int b = blockIdx.x * 8 + w;
  float*       xr = X  + (size_t)b * CPAD + l * 4;
  const float* ar = AX + (size_t)b * CPAD + l * 4;
  float x4[4], num[4];
  float s = 0.0f;
  #pragma unroll
  for (int j = 0; j < 4; ++j) {
    x4[j]  = xr[j];
    num[j] = x4[j] * ar[j];
    s += num[j];
  }
  #pragma unroll
  for (int m = 16; m >= 1; m >>= 1) s += __shfl_xor(s, m, 32);
  float inv = 1.0f / (s + 1e-8f);
  float ent = 0.0f;
  #pragma unroll
  for (int j = 0; j < 4; ++j) {
    float d = num[j] * inv;
    ent += d * logf(d + 1e-20f);
    xr[j] = x4[j] + d;
    int c = l * 4 + j;
    if (c < NCLS) outX[(size_t)b * (NCLS * ITERS) + c * ITERS + it] = d;
  }
  #pragma unroll
  for (int m = 16; m >= 1; m >>= 1) ent += __shfl_xor(ent, m, 32);
  if (l == 0) outEnt[(size_t)b * ITERS + it] = -ent;
}

// ----------------------------------------------------------------
extern "C" void kernel_launch(void* const* d_in, const int* in_sizes, int n_in,
                              void* d_out, int out_size, void* d_ws, size_t ws_size,
                              hipStream_t stream) {
  (void)in_sizes; (void)n_in; (void)out_size; (void)ws_size;
  const float* emb = (const float*)d_in[0];
  const int*   lab = (const int*)d_in[1];
  float* ws  = (float*)d_ws;
  float* ne  = ws + NE_OFF;
  float* A   = ws + A_OFF;
  float* X   = ws + X_OFF;
  float* AX  = ws + AX_OFF;
  float* sum = ws + SUM_OFF;
  float* outX   = (float*)d_out;                              // [4096,100,30]
  float* outEnt = (float*)d_out + (size_t)BSZ * NCLS * ITERS; // [4096,30]

  init_x<<<(BSZ * CPAD) / 256, 256, 0, stream>>>(lab, X, sum);
  normalize_rows<<<BSZ, 256, 0, stream>>>(emb, ne);
  gemm_syrk<<<dim3(BSZ / 128, BSZ / 64), 256, 0, stream>>>(ne, A, sum);
  threshold_mean<<<(BSZ * (size_t)BSZ) / (256 * 4), 256, 0, stream>>>(A, sum);
  for (int it = 0; it < ITERS; ++it) {
    gemm_ax<<<BSZ / 32, 256, 0, stream>>>(A, X, AX);
    replicator_step<<<BSZ / 8, 256, 0, stream>>>(AX, X, outX, outEnt, it);
  }
}